// SRA_LSTM_16716012716120
// MI455X (gfx1250) — compile-verified
//
#include <hip/hip_runtime.h>
#include <math.h>

typedef __attribute__((ext_vector_type(2))) float v2f;
typedef __attribute__((ext_vector_type(8))) float v8f;

#define PED   512
#define HDIM  64
#define EDIM  32
#define KT    24          // K tiles: 96 / 4
#define NT    16          // N tiles: 256 / 16
#define NGATE 256         // 4*HDIM

__device__ __forceinline__ float fast_sigmoid(float x) {
    return __frcp_rn(1.0f + __expf(-x));
}
__device__ __forceinline__ float fast_tanh(float x) {
    // tanh(x) = 1 - 2/(1 + e^{2x}); saturates correctly for |x| large.
    return fmaf(-2.0f, __frcp_rn(1.0f + __expf(2.0f * x)), 1.0f);
}

// One wave handles a 16-row block of pairs and half of the hidden units
// (tHalf selects hidden tiles t = 2*tHalf .. 2*tHalf+1). For each t, the
// accumulators q=0..3 hold gate tiles i/f/g/o at gate columns nt = t + 4q,
// so LSTM elementwise math happens in-register at a fixed (vgpr, lane) slot.
__global__ __launch_bounds__(256)
void sra_lstm_wmma_kernel(const float* __restrict__ corr_index,
                          const float* __restrict__ rela_ht,
                          const float* __restrict__ rela_ct,
                          const int*   __restrict__ nei_index,
                          const float* __restrict__ W_emb,
                          const float* __restrict__ b_emb,
                          const float* __restrict__ W_ih,
                          const float* __restrict__ W_hh,
                          const float* __restrict__ b_ih,
                          const float* __restrict__ b_hh,
                          float* __restrict__ h_out,
                          float* __restrict__ c_out)
{
    // B fragments pre-swizzled into WMMA lane order:
    // Wfrag[((kt*NT + nt)*32 + lane)*2 + j] = Wc^T[kt*4 + 2*(lane/16) + j][nt*16 + lane%16]
    // -> each fragment is one aligned ds_load_b64, bank-conflict free.
    __shared__ float Wfrag[KT * NT * 64];        // 96 KB
    __shared__ float biasLds[NGATE];
    __shared__ float WembLds[EDIM * 2];
    __shared__ float bembLds[EDIM];

    {
        #pragma unroll 4
        for (int i = 0; i < (KT * NT * 64) / 256; ++i) {   // 96 dwords/thread
            const int d    = i * 256 + threadIdx.x;
            const int j    = d & 1;
            const int lane = (d >> 1) & 31;
            const int nt   = (d >> 6) & 15;
            const int kt   = d >> 10;
            const int col  = nt * 16 + (lane & 15);        // gate column n
            const int k    = kt * 4 + ((lane >> 4) << 1) + j;
            Wfrag[d] = (k < EDIM) ? W_ih[col * EDIM + k]
                                  : W_hh[col * HDIM + (k - EDIM)];
        }
        const int n = threadIdx.x;
        biasLds[n] = b_ih[n] + b_hh[n];
        if (n < EDIM * 2) WembLds[n] = W_emb[n];
        if (n < EDIM)     bembLds[n] = b_emb[n];
    }
    __syncthreads();

    const int lane     = threadIdx.x & 31;
    const int wave     = threadIdx.x >> 5;       // 0..7
    const int rowBlock = blockIdx.x * 4 + (wave >> 1);
    const int tHalf    = wave & 1;
    const int rowBase  = rowBlock * 16;
    const int mrow     = lane & 15;              // M (A) / N (B,C,D) index
    const int half     = lane >> 4;

    // ---- A operand per lane: X[m][k], m = rowBase+mrow,
    //      lane holds k with (k mod 4) in {2*half, 2*half+1}  -> 48 values.
    float aX[48];
    const int pair = rowBase + mrow;
    {
        const float* hrow = rela_ht + (size_t)pair * HDIM;
        #pragma unroll
        for (int t = 0; t < 16; ++t) {           // h part: k = 32 + 4t + 2*half + j
            v2f v = *(const v2f*)(hrow + t * 4 + 2 * half);
            aX[16 + 2 * t]     = v.x;
            aX[16 + 2 * t + 1] = v.y;
        }
        const float x0 = corr_index[(size_t)pair * 2 + 0];
        const float x1 = corr_index[(size_t)pair * 2 + 1];
        #pragma unroll
        for (int t = 0; t < 8; ++t) {            // emb part: k = 4t + 2*half + j
            #pragma unroll
            for (int j = 0; j < 2; ++j) {
                const int e = t * 4 + 2 * half + j;
                float v = fmaf(x0, WembLds[e * 2 + 0],
                          fmaf(x1, WembLds[e * 2 + 1], bembLds[e]));
                aX[2 * t + j] = fmaxf(v, 0.0f);  // ReLU
            }
        }
    }

    // ---- GEMM: 24 K-steps of V_WMMA_F32_16X16X4_F32; first step uses C=0.
    v8f acc[2][4];
    v8f zacc;
    #pragma unroll
    for (int r = 0; r < 8; ++r) zacc[r] = 0.0f;

    const float* fb = Wfrag + (lane << 1);
    #pragma unroll
    for (int kt = 0; kt < KT; ++kt) {
        v2f a;
        a.x = aX[2 * kt];
        a.y = aX[2 * kt + 1];
        #pragma unroll
        for (int tt = 0; tt < 2; ++tt) {
            #pragma unroll
            for (int q = 0; q < 4; ++q) {
                const int nt = (tHalf * 2 + tt) + q * 4;
                const v2f b = *(const v2f*)(fb + ((kt * NT + nt) << 6));
                acc[tt][q] = __builtin_amdgcn_wmma_f32_16x16x4_f32(
                    false, a, false, b, (short)0,
                    (kt == 0) ? zacc : acc[tt][q], false, false);
            }
        }
    }

    // ---- LSTM elementwise + masked scatter (bias folded in here).
    #pragma unroll
    for (int tt = 0; tt < 2; ++tt) {
        const int t    = tHalf * 2 + tt;
        const int jcol = t * 16 + mrow;          // hidden unit index 0..63
        const float bI = biasLds[jcol];
        const float bF = biasLds[jcol + 64];
        const float bG = biasLds[jcol + 128];
        const float bO = biasLds[jcol + 192];
        #pragma unroll
        for (int r = 0; r < 8; ++r) {
            const int m = rowBase + r + 8 * half;
            const size_t idx = (size_t)m * HDIM + jcol;
            const float iv = fast_sigmoid(acc[tt][0][r] + bI);
            const float fv = fast_sigmoid(acc[tt][1][r] + bF);
            const float gv = fast_tanh   (acc[tt][2][r] + bG);
            const float ov = fast_sigmoid(acc[tt][3][r] + bO);
            const float cold = rela_ct[idx];
            const float hold = rela_ht[idx];
            const float cnew = fmaf(fv, cold, iv * gv);
            const float hnew = ov * fast_tanh(cnew);
            const bool msk = nei_index[m] > 0;
            h_out[idx] = msk ? hnew : hold;
            c_out[idx] = msk ? cnew : cold;
        }
    }
}

extern "C" void kernel_launch(void* const* d_in, const int* in_sizes, int n_in,
                              void* d_out, int out_size, void* d_ws, size_t ws_size,
                              hipStream_t stream)
{
    (void)in_sizes; (void)n_in; (void)d_ws; (void)ws_size; (void)out_size;
    const float* corr_index = (const float*)d_in[0];
    const float* rela_ht    = (const float*)d_in[1];
    const float* rela_ct    = (const float*)d_in[2];
    const int*   nei_index  = (const int*)  d_in[3];
    const float* W_emb      = (const float*)d_in[4];
    const float* b_emb      = (const float*)d_in[5];
    const float* W_ih       = (const float*)d_in[6];
    const float* W_hh       = (const float*)d_in[7];
    const float* b_ih       = (const float*)d_in[8];
    const float* b_hh       = (const float*)d_in[9];

    float* h_out = (float*)d_out;
    float* c_out = h_out + (size_t)PED * PED * HDIM;

    // 16384 row-blocks of 16 pairs; 2 waves per row-block; 8 waves per block.
    const int grid = (PED * PED / 16) / 4;   // 4096 blocks
    sra_lstm_wmma_kernel<<<grid, 256, 0, stream>>>(
        corr_index, rela_ht, rela_ct, nei_index,
        W_emb, b_emb, W_ih, W_hh, b_ih, b_hh, h_out, c_out);
}